// MultiHeadAttentionBern_61074434949245
// MI455X (gfx1250) — compile-verified
//
#include <hip/hip_runtime.h>
#include <math.h>

// ---------------------------------------------------------------------------
// MultiHeadAttention with Bernoulli (Gumbel hard) gating — MI455X / gfx1250
// Bandwidth-bound (268MB gumbel read dominates) -> fp32 WMMA everywhere.
// v2: transposed q/k workspace for coalesced WMMA fragment loads, padded LDS
//     (no bank conflicts), hoisted q fragments, dual accumulators for ILP,
//     LDS-staged X tile in the projection GEMM.
// ---------------------------------------------------------------------------

typedef __attribute__((ext_vector_type(2))) float v2f;
typedef __attribute__((ext_vector_type(8))) float v8f;

#define BB   4
#define SS   1024
#define EE   512
#define HH   8
#define DKK  64

__device__ __forceinline__ v8f wmma_f32_4(v2f a, v2f b, v8f c) {
  // V_WMMA_F32_16X16X4_F32 : D = A(16x4) * B(4x16) + C(16x16), all f32
  return __builtin_amdgcn_wmma_f32_16x16x4_f32(
      /*neg_a=*/false, a, /*neg_b=*/false, b,
      /*c_mod=*/(short)0, c, /*reuse_a=*/false, /*reuse_b=*/false);
}

// ---------------------------------------------------------------------------
// C[4096,512] = X[4096,512] @ W[512,512] + bias.
// mode 0: flat [4096,512]        (output projection)
// mode 1: head-split [bh][s][dk] (v)
// mode 2: transposed  [bh][dk][s] (q, k -> coalesced attn fragment loads)
// block = 128 threads (4 waves); each wave owns one 16x16 tile;
// workgroup covers 16 rows x 64 cols. grid = (256, 8).
// ---------------------------------------------------------------------------
__global__ __launch_bounds__(128)
void gemm_proj(const float* __restrict__ X, const float* __restrict__ W,
               const float* __restrict__ bias, float* __restrict__ out,
               int mode) {
  __shared__ float Xs[16][EE + 1];          // padded: bank = (row + k) % 64

  const int tid  = threadIdx.x;
  const int lane = tid & 31;
  const int wave = tid >> 5;
  const int half = (lane < 16) ? 0 : 1;     // lane-half: K pair / row+8
  const int l15  = lane & 15;
  const int r0 = blockIdx.x * 16;
  const int c0 = blockIdx.y * 64 + wave * 16;

  // stage X tile (coalesced: consecutive threads -> consecutive columns)
  for (int i = tid; i < 16 * EE; i += 128) {
    const int r = i >> 9, c = i & (EE - 1);
    Xs[r][c] = X[(size_t)(r0 + r) * EE + c];
  }
  __syncthreads();

  v8f acc0 = {0.f,0.f,0.f,0.f,0.f,0.f,0.f,0.f};
  v8f acc1 = {0.f,0.f,0.f,0.f,0.f,0.f,0.f,0.f};
  for (int kk = 0; kk < EE; kk += 8) {      // two WMMA per iter -> ILP
    const int ka0 = kk + half * 2;
    const int ka1 = kk + 4 + half * 2;
    v2f a0, b0, a1, b1;
    a0.x = Xs[l15][ka0 + 0];  a0.y = Xs[l15][ka0 + 1];
    b0.x = W[(size_t)(ka0 + 0) * EE + c0 + l15];
    b0.y = W[(size_t)(ka0 + 1) * EE + c0 + l15];
    a1.x = Xs[l15][ka1 + 0];  a1.y = Xs[l15][ka1 + 1];
    b1.x = W[(size_t)(ka1 + 0) * EE + c0 + l15];
    b1.y = W[(size_t)(ka1 + 1) * EE + c0 + l15];
    acc0 = wmma_f32_4(a0, b0, acc0);
    acc1 = wmma_f32_4(a1, b1, acc1);
  }

  // C/D layout: VGPR i holds rows (i, i+8); lanes 0-15 / 16-31
  #pragma unroll
  for (int i = 0; i < 8; ++i) {
    const int row = r0 + i + half * 8;
    const int col = c0 + l15;
    const float val = acc0[i] + acc1[i] + bias[col];
    if (mode == 0) {
      out[(size_t)row * EE + col] = val;
    } else {
      const int b_ = row >> 10, s_ = row & (SS - 1);
      const int h_ = col >> 6,  d_ = col & (DKK - 1);
      const int bh = b_ * HH + h_;
      if (mode == 1) out[((size_t)bh * SS + s_) * DKK + d_] = val;
      else           out[((size_t)bh * DKK + d_) * SS + s_] = val;
    }
  }
}

// ---------------------------------------------------------------------------
// Attention core. grid = (S/16 = 64, B = 4), block = 128 (4 waves).
// A workgroup owns (batch b, 16 query rows) across ALL 8 heads, so adj
// accumulates in LDS and is written once (no atomics).
// qT,kT layout: [bh][dk][s]; v layout: [bh][s][dk].
// ---------------------------------------------------------------------------
__global__ __launch_bounds__(128)
void attn(const float* __restrict__ qT, const float* __restrict__ kT,
          const float* __restrict__ v, const float* __restrict__ gum,
          float* __restrict__ merged, float* __restrict__ adj) {
  __shared__ float Lt[16][SS + 1];   // logits -> masked probs (padded rows)
  __shared__ float adjacc[16][SS];   // adj accumulator
  __shared__ float red[16][8];
  __shared__ float rowmax[16];
  __shared__ float rowsum[16];

  const int tid  = threadIdx.x;
  const int lane = tid & 31;
  const int wave = tid >> 5;
  const int half = (lane < 16) ? 0 : 1;
  const int l15  = lane & 15;
  const int b_   = blockIdx.y;
  const int s0   = blockIdx.x * 16;   // first query row of this tile

  for (int i = tid; i < 16 * SS; i += 128) (&adjacc[0][0])[i] = 0.f;
  __syncthreads();

  for (int h = 0; h < HH; ++h) {
    const int bh = b_ * HH + h;
    const float* qh = qT + (size_t)bh * DKK * SS;   // [dk][s]
    const float* kh = kT + (size_t)bh * DKK * SS;   // [dk][s]
    const float* vh = v  + (size_t)bh * SS * DKK;   // [s][dk]

    // ---- Phase A: logits tile [16 x 1024] = q_tile @ k^T * (1/sqrt(64))
    // Hoist q fragments (ct-invariant): 16 x v2f = 32 VGPRs.
    v2f qa[16];
    #pragma unroll
    for (int kk4 = 0; kk4 < 16; ++kk4) {
      const int ka = kk4 * 4 + half * 2;
      qa[kk4].x = qh[(size_t)(ka + 0) * SS + s0 + l15];  // coalesced
      qa[kk4].y = qh[(size_t)(ka + 1) * SS + s0 + l15];
    }
    for (int ct = wave; ct < SS / 16; ct += 4) {     // uniform: 16 iters/wave
      const int t0 = ct * 16;
      v8f acc0 = {0.f,0.f,0.f,0.f,0.f,0.f,0.f,0.f};
      v8f acc1 = {0.f,0.f,0.f,0.f,0.f,0.f,0.f,0.f};
      #pragma unroll
      for (int kk4 = 0; kk4 < 16; kk4 += 2) {
        const int ka0 = kk4 * 4 + half * 2;
        const int ka1 = ka0 + 4;
        v2f b0, b1;
        b0.x = kh[(size_t)(ka0 + 0) * SS + t0 + l15];    // coalesced
        b0.y = kh[(size_t)(ka0 + 1) * SS + t0 + l15];
        b1.x = kh[(size_t)(ka1 + 0) * SS + t0 + l15];
        b1.y = kh[(size_t)(ka1 + 1) * SS + t0 + l15];
        acc0 = wmma_f32_4(qa[kk4 + 0], b0, acc0);
        acc1 = wmma_f32_4(qa[kk4 + 1], b1, acc1);
      }
      #pragma unroll
      for (int i = 0; i < 8; ++i)
        Lt[i + half * 8][t0 + l15] = (acc0[i] + acc1[i]) * 0.125f;
    }
    __syncthreads();

    // ---- Phase B: per-row softmax stats (16 rows x 8 threads each)
    {
      const int r = tid >> 3, j = tid & 7;
      float m = -1e30f;
      for (int t = j; t < SS; t += 8) m = fmaxf(m, Lt[r][t]);
      red[r][j] = m;
      __syncthreads();
      if (j == 0) {
        float mm = red[r][0];
        for (int p = 1; p < 8; ++p) mm = fmaxf(mm, red[r][p]);
        rowmax[r] = mm;
      }
      __syncthreads();
      const float mm = rowmax[r];
      float sum = 0.f;
      for (int t = j; t < SS; t += 8) sum += __expf(Lt[r][t] - mm);
      red[r][j] = sum;
      __syncthreads();
      if (j == 0) {
        float ss = 0.f;
        for (int p = 0; p < 8; ++p) ss += red[r][p];
        rowsum[r] = ss;
      }
      __syncthreads();
    }

    // ---- Phase C: gumbel hard gate + probs, rewrite Lt in place, adj +=
    // t is fastest across threads -> coalesced 8B/lane gumbel loads.
    const float2* gum2 = (const float2*)gum;
    for (int i = tid; i < 16 * SS; i += 128) {
      const int r = i >> 10, t = i & (SS - 1);
      const float l = Lt[r][t];
      const float p = __expf(l - rowmax[r]) / rowsum[r];
      const float2 g = gum2[((size_t)bh * SS + (s0 + r)) * SS + t];
      // hard gate: logits + gumbel(g.y) - gumbel(g.x) > 0 (tau-independent)
      const float d = l - __logf(-__logf(g.y)) + __logf(-__logf(g.x));
      const float a = (d > 0.f) ? 1.f : 0.f;
      Lt[r][t] = a * p;
      adjacc[r][t] += a * (1.0f / HH);
    }
    __syncthreads();

    // ---- Phase D: hidden[16 x 64] = (A*probs)[16x1024] @ v[1024x64]
    {
      const int c0 = wave * 16;                      // each wave: 16 v-cols
      v8f acc0 = {0.f,0.f,0.f,0.f,0.f,0.f,0.f,0.f};
      v8f acc1 = {0.f,0.f,0.f,0.f,0.f,0.f,0.f,0.f};
      for (int kk = 0; kk < SS; kk += 8) {
        const int ka0 = kk + half * 2;
        const int ka1 = kk + 4 + half * 2;
        v2f a0, b0, a1, b1;
        a0.x = Lt[l15][ka0 + 0];                     // padded: conflict-free
        a0.y = Lt[l15][ka0 + 1];
        a1.x = Lt[l15][ka1 + 0];
        a1.y = Lt[l15][ka1 + 1];
        b0.x = vh[(size_t)(ka0 + 0) * DKK + c0 + l15];
        b0.y = vh[(size_t)(ka0 + 1) * DKK + c0 + l15];
        b1.x = vh[(size_t)(ka1 + 0) * DKK + c0 + l15];
        b1.y = vh[(size_t)(ka1 + 1) * DKK + c0 + l15];
        acc0 = wmma_f32_4(a0, b0, acc0);
        acc1 = wmma_f32_4(a1, b1, acc1);
      }
      #pragma unroll
      for (int i = 0; i < 8; ++i) {
        const int srow = s0 + i + half * 8;
        const int col  = h * DKK + c0 + l15;         // merged head layout
        merged[((size_t)b_ * SS + srow) * EE + col] = acc0[i] + acc1[i];
      }
    }
    __syncthreads();   // Lt reused next head
  }

  // ---- write adj tile (coalesced)
  for (int i = tid; i < 16 * SS; i += 128) {
    const int r = i >> 10, t = i & (SS - 1);
    adj[((size_t)b_ * SS + (s0 + r)) * SS + t] = adjacc[r][t];
  }
}

// ---------------------------------------------------------------------------
extern "C" void kernel_launch(void* const* d_in, const int* in_sizes, int n_in,
                              void* d_out, int out_size, void* d_ws, size_t ws_size,
                              hipStream_t stream) {
  const float* queries = (const float*)d_in[0];
  const float* keys    = (const float*)d_in[1];
  const float* values  = (const float*)d_in[2];
  const float* Wq = (const float*)d_in[3];
  const float* bq = (const float*)d_in[4];
  const float* Wk = (const float*)d_in[5];
  const float* bk = (const float*)d_in[6];
  const float* Wv = (const float*)d_in[7];
  const float* bv = (const float*)d_in[8];
  const float* Wo = (const float*)d_in[9];
  const float* bo = (const float*)d_in[10];
  const float* gum = (const float*)d_in[11];

  const size_t mat = (size_t)BB * SS * EE;   // 2,097,152 floats (8 MB)
  float* q_ws   = (float*)d_ws;              // [bh][dk][s] (transposed)
  float* k_ws   = q_ws + mat;                // [bh][dk][s] (transposed)
  float* v_ws   = k_ws + mat;                // [bh][s][dk]
  float* merged = v_ws + mat;                // [b][s][e]; total ws: 32 MB

  float* out_p = (float*)d_out;              // [B,S,E]
  float* adj_p = out_p + mat;                // [B,S,S]

  dim3 gblk(128);
  dim3 ggrd(SS * BB / 16, EE / 64);          // (256, 8)

  gemm_proj<<<ggrd, gblk, 0, stream>>>(queries, Wq, bq, q_ws, 2);
  gemm_proj<<<ggrd, gblk, 0, stream>>>(keys,    Wk, bk, k_ws, 2);
  gemm_proj<<<ggrd, gblk, 0, stream>>>(values,  Wv, bv, v_ws, 1);

  attn<<<dim3(SS / 16, BB), gblk, 0, stream>>>(q_ws, k_ws, v_ws, gum,
                                               merged, adj_p);

  gemm_proj<<<ggrd, gblk, 0, stream>>>(merged, Wo, bo, out_p, 0);
}